// FullModel_88115549045301
// MI455X (gfx1250) — compile-verified
//
#include <hip/hip_runtime.h>
#include <math.h>

// ---------------------------------------------------------------------------
// MI455X / gfx1250 fused kernel for the 1D-conv network.
//   * 1 workgroup (256 thr = 8 wave32) per 16 samples; whole rows in LDS.
//   * Branch A (5 circular convs) + Branch B conv1 run on the matrix pipe
//     (V_WMMA_F32_16X16X4_F32, exact fp32):
//        A[r][k] = w[k-r+p]  (Toeplitz weights, cached in VGPRs per layer)
//        B[k][c] = y[c][i0+k] (16 samples, transposed stride-17 in LDS)
//     Physical wrap halos on each row buffer remove all per-step address
//     math: steady state is 2 ds_load (immediate offsets) + 1 wmma per step.
//   * Dynamic per-sample conv (k=51) is depthwise -> VALU, register-blocked:
//     51 taps + 55-wide window in VGPRs, 5 outputs/block, zeroed halo.
//   * LDS: 2x (2144 rows x 17) ping-pong + weight pads + f = 298,944 B.
// ---------------------------------------------------------------------------

typedef __attribute__((ext_vector_type(2))) float v2f;
typedef __attribute__((ext_vector_type(8))) float v8f;

#define NB    8192
#define LEN   2000
#define NS    16
#define STR   17            // padded LDS stride (floats) per position
#define NT    256
#define NW    8
#define NTILE (LEN / 16)    // 125
#define HL    64            // left halo (>= 60 needed for k=121)
#define HR    80            // right halo (>= 60 needed)
#define ROWS  (HL + LEN + HR)       // 2144
#define BUFF  (ROWS * STR)          // 36448 floats per buffer

// LDS layout (float offsets)
#define OFF_BUF0  0
#define OFF_BUF1  BUFF
#define OFF_WPAD  (2 * BUFF)                 // 72896
#define WSLOT     160                        // per-layer zero-padded weights
#define OFF_WPB   (OFF_WPAD + 5 * WSLOT)     // 73696, branch-B conv1 pad
#define WPB_N     224
#define OFF_F     (OFF_WPB + WPB_N)          // 73920 : f_t[51][16]
#define SMEMF     (OFF_F + 51 * NS)          // 74736 floats = 298,944 B

__device__ __forceinline__ float lrelu(float v) {
  return v > 0.f ? v : 0.01f * v;
}

// Copy wrapped halo of a row buffer (buf points at pos 0).
__device__ __forceinline__ void fill_wrap_halo(float* buf, int tid) {
  for (int idx = tid; idx < (HL + HR) * NS; idx += NT) {
    const int c = idx & (NS - 1);
    const int h = idx >> 4;                       // 0..143
    const int pos  = (h < HL) ? (h - HL) : (LEN + (h - HL));
    const int srcp = (pos < 0) ? pos + LEN : pos - LEN;
    buf[pos * STR + c] = buf[srcp * STR + c];
  }
}

// Zero +-32 halo (for the zero-padded dynamic conv).
__device__ __forceinline__ void fill_zero_halo(float* buf, int tid) {
  for (int idx = tid; idx < 64 * NS; idx += NT) {
    const int c = idx & (NS - 1);
    const int h = idx >> 4;                       // 0..63
    const int pos = (h < 32) ? (h - 32) : (LEN + (h - 32));
    buf[pos * STR + c] = 0.f;
  }
}

// One Branch-A circular conv layer on the matrix pipe.
// dst[c][i] = lrelu(b + sum_m w[m]*src[c][(i+m-p) mod L]) + src[c][i]
template <int KW>
__device__ __forceinline__ void convA_layer(const float* __restrict__ wslot,
                                            float bb,
                                            const float* __restrict__ src,
                                            float* __restrict__ dst,
                                            int wave, int lane) {
  constexpr int P     = KW / 2;
  constexpr int STEPS = (KW + 15 + 3) / 4;        // ceil((KW+15)/4)
  const int M16  = lane & 15;
  const int half = lane >> 4;
  const int k0   = 2 * half;                      // K of VGPR0 per ISA layout

  // Tile-invariant Toeplitz A fragments -> registers (once per layer).
  v2f afr[STEPS];
#pragma unroll
  for (int s = 0; s < STEPS; ++s) {
    afr[s].x = wslot[16 + 4 * s + k0 - M16];      // w[(kb+K)-r+p], p cancels
    afr[s].y = wslot[16 + 4 * s + k0 + 1 - M16];
  }

  for (int t = wave; t < NTILE; t += NW) {        // uniform per wave
    const int i0 = t * 16;
    v8f acc = {};
    const float* bbase = src + (i0 - P + k0) * STR + M16;  // halo-covered
#pragma unroll
    for (int s = 0; s < STEPS; ++s) {
      v2f b;                                      // immediate DS offsets
      b.x = bbase[(4 * s) * STR];
      b.y = bbase[(4 * s) * STR + STR];
      acc = __builtin_amdgcn_wmma_f32_16x16x4_f32(
          false, afr[s], false, b, (short)0, acc, false, false);
    }
#pragma unroll
    for (int j = 0; j < 8; ++j) {                 // D: M = vgpr + 8*laneHalf
      const int pos = i0 + j + 8 * half;
      const float prev = src[pos * STR + M16];
      dst[pos * STR + M16] = lrelu(acc[j] + bb) + prev;
    }
  }
}

// Branch-B conv1 (k=100, stride 4, VALID, tanh) on the matrix pipe.
// A[r][k] = wB1[k-4r], k in [0,160): 40 K-steps, 30 tiles of 16 outputs.
__device__ __forceinline__ void convB1_wmma(const float* __restrict__ wpB,
                                            float bb,
                                            const float* __restrict__ xb,
                                            float* __restrict__ t1,
                                            int wave, int lane) {
  const int M16  = lane & 15;
  const int half = lane >> 4;
  const int k0   = 2 * half;
  v2f afr[40];
#pragma unroll
  for (int s = 0; s < 40; ++s) {
    afr[s].x = wpB[64 + 4 * s + k0 - 4 * M16];
    afr[s].y = wpB[64 + 4 * s + k0 + 1 - 4 * M16];
  }
  for (int t = wave; t < 30; t += NW) {
    const int o0 = t * 16;
    v8f acc = {};
    const float* bbase = xb + (4 * o0 + k0) * STR + M16;   // max pos 2015 < halo end
#pragma unroll
    for (int s = 0; s < 40; ++s) {
      v2f b;
      b.x = bbase[(4 * s) * STR];
      b.y = bbase[(4 * s) * STR + STR];
      acc = __builtin_amdgcn_wmma_f32_16x16x4_f32(
          false, afr[s], false, b, (short)0, acc, false, false);
    }
#pragma unroll
    for (int j = 0; j < 8; ++j) {
      const int o = o0 + j + 8 * half;
      if (o < 476) t1[o * NS + M16] = tanhf(acc[j] + bb);
    }
  }
}

__launch_bounds__(NT, 1)
__global__ void fused_conv_net(
    const float* __restrict__ x,
    const float* __restrict__ wA1, const float* __restrict__ bA1,
    const float* __restrict__ wA2, const float* __restrict__ bA2,
    const float* __restrict__ wA3, const float* __restrict__ bA3,
    const float* __restrict__ wA4, const float* __restrict__ bA4,
    const float* __restrict__ wA5, const float* __restrict__ bA5,
    const float* __restrict__ wB1, const float* __restrict__ bB1,
    const float* __restrict__ wB2, const float* __restrict__ bB2,
    const float* __restrict__ wC,  const float* __restrict__ bC,
    float* __restrict__ out)
{
  extern __shared__ float smem[];
  float* buf0 = smem + OFF_BUF0 + HL * STR;   // pos-0-based, halo below/above
  float* buf1 = smem + OFF_BUF1 + HL * STR;
  float* fT   = smem + OFF_F;
  float* t1   = smem + OFF_BUF1;              // alias: dead once layer 1 runs
  float* t2   = t1 + 476 * NS;

  const int tid  = threadIdx.x;
  const int lane = tid & 31;                  // wave32
  const int wave = tid >> 5;
  const size_t base = (size_t)blockIdx.x * NS * LEN;

  // ---- 1) Load 16 rows of x, transposed buf0[pos*17 + c] -------------------
  for (int idx = tid; idx < NS * LEN; idx += NT) {
    const int c = idx / LEN;
    const int i = idx - c * LEN;              // consecutive lanes: coalesced
    buf0[i * STR + c] = x[base + idx];
  }
  __syncthreads();

  // ---- 2) x wrap halo + all zero-padded weight slots (single pass) ---------
  fill_wrap_halo(buf0, tid);
  {
    const float* ws[5] = {wA1, wA2, wA3, wA4, wA5};
    const int    kk[5] = {121, 101, 51, 31, 21};
#pragma unroll
    for (int l = 0; l < 5; ++l) {
      float* wp = smem + OFF_WPAD + l * WSLOT;
      for (int i = tid; i < WSLOT; i += NT) {
        const int j = i - 16;
        wp[i] = (j >= 0 && j < kk[l]) ? ws[l][j] : 0.f;
      }
    }
    float* wpB = smem + OFF_WPB;
    for (int i = tid; i < WPB_N; i += NT) {
      const int j = i - 64;
      wpB[i] = (j >= 0 && j < 100) ? wB1[j] : 0.f;
    }
  }
  __syncthreads();

  // ---- 3) Branch B: conv1 (WMMA) -> pool4 -> conv2 -> f_t[51][16] ----------
  convB1_wmma(smem + OFF_WPB, bB1[0], buf0, t1, wave, lane);
  __syncthreads();
  for (int idx = tid; idx < NS * 119; idx += NT) {
    const int c = idx & (NS - 1), q = idx >> 4;
    float m = t1[(4 * q) * NS + c];
    m = fmaxf(m, t1[(4 * q + 1) * NS + c]);
    m = fmaxf(m, t1[(4 * q + 2) * NS + c]);
    m = fmaxf(m, t1[(4 * q + 3) * NS + c]);
    t2[q * NS + c] = m;
  }
  __syncthreads();
  {
    const float bb = bB2[0];
    for (int idx = tid; idx < NS * 51; idx += NT) {
      const int c = idx & (NS - 1), u = idx >> 4;
      float s = bb;
#pragma unroll
      for (int j = 0; j < 19; ++j) s += wB2[j] * t2[(2 * u + j) * NS + c];
      fT[u * NS + c] = tanhf(s);
    }
  }
  __syncthreads();

  // ---- 4) Branch A: 5 WMMA conv layers, ping-pong with wrap halos ----------
  const float* wp = smem + OFF_WPAD;
  convA_layer<121>(wp + 0 * WSLOT, bA1[0], buf0, buf1, wave, lane);
  __syncthreads(); fill_wrap_halo(buf1, tid); __syncthreads();
  convA_layer<101>(wp + 1 * WSLOT, bA2[0], buf1, buf0, wave, lane);
  __syncthreads(); fill_wrap_halo(buf0, tid); __syncthreads();
  convA_layer<51>(wp + 2 * WSLOT, bA3[0], buf0, buf1, wave, lane);
  __syncthreads(); fill_wrap_halo(buf1, tid); __syncthreads();
  convA_layer<31>(wp + 3 * WSLOT, bA4[0], buf1, buf0, wave, lane);
  __syncthreads(); fill_wrap_halo(buf0, tid); __syncthreads();
  convA_layer<21>(wp + 4 * WSLOT, bA5[0], buf0, buf1, wave, lane);
  __syncthreads(); fill_zero_halo(buf1, tid); __syncthreads();   // y := buf1

  // ---- 5) Dynamic per-sample conv (k=51, zero pad) + merge = dyn*y ---------
  // Register-blocked: 51 taps + 55-wide window in VGPRs, 5 outputs/block.
  {
    float* y  = buf1;
    float* mg = buf0;
    const int c = tid & (NS - 1);
    const int ibase = (tid >> 4) * 125;       // contiguous 125-wide segment
    float fr[51];
#pragma unroll
    for (int u = 0; u < 51; ++u) fr[u] = fT[u * NS + c];
    for (int blk = 0; blk < 25; ++blk) {
      const int i0 = ibase + blk * 5;
      float w[55];
#pragma unroll
      for (int q = 0; q < 55; ++q) w[q] = y[(i0 - 25 + q) * STR + c];
#pragma unroll
      for (int v = 0; v < 5; ++v) {
        float s = 0.f;
#pragma unroll
        for (int u = 0; u < 51; ++u) s += fr[u] * w[v + u];
        mg[(i0 + v) * STR + c] = s * w[v + 25];
      }
    }
  }
  __syncthreads();

  // tiny wrap halo (+-2) for the final circular conv
  {
    float* mg = buf0;
    for (int idx = tid; idx < 4 * NS; idx += NT) {
      const int c = idx & (NS - 1);
      const int h = idx >> 4;                 // 0..3
      const int pos  = (h < 2) ? (h - 2) : (LEN + (h - 2));
      const int srcp = (pos < 0) ? pos + LEN : pos - LEN;
      mg[pos * STR + c] = mg[srcp * STR + c];
    }
  }
  __syncthreads();

  // ---- 6) Final circular conv k=5 + |.|, coalesced store -------------------
  {
    const float* mg = buf0;
    const float bc = bC[0];
    const float c0 = wC[0], c1 = wC[1], c2 = wC[2], c3 = wC[3], c4 = wC[4];
    for (int idx = tid; idx < NS * LEN; idx += NT) {
      const int c = idx / LEN;
      const int i = idx - c * LEN;
      const float* b = mg + i * STR + c;
      float s = bc;
      s += c0 * b[-2 * STR];
      s += c1 * b[-STR];
      s += c2 * b[0];
      s += c3 * b[STR];
      s += c4 * b[2 * STR];
      out[base + idx] = fabsf(s);
    }
  }
}

extern "C" void kernel_launch(void* const* d_in, const int* in_sizes, int n_in,
                              void* d_out, int out_size, void* d_ws, size_t ws_size,
                              hipStream_t stream) {
  (void)in_sizes; (void)n_in; (void)out_size; (void)d_ws; (void)ws_size;
  const float* x   = (const float*)d_in[0];
  const float* wA1 = (const float*)d_in[1];  const float* bA1 = (const float*)d_in[2];
  const float* wA2 = (const float*)d_in[3];  const float* bA2 = (const float*)d_in[4];
  const float* wA3 = (const float*)d_in[5];  const float* bA3 = (const float*)d_in[6];
  const float* wA4 = (const float*)d_in[7];  const float* bA4 = (const float*)d_in[8];
  const float* wA5 = (const float*)d_in[9];  const float* bA5 = (const float*)d_in[10];
  const float* wB1 = (const float*)d_in[11]; const float* bB1 = (const float*)d_in[12];
  const float* wB2 = (const float*)d_in[13]; const float* bB2 = (const float*)d_in[14];
  const float* wC  = (const float*)d_in[15]; const float* bC  = (const float*)d_in[16];
  float* outp = (float*)d_out;

  const size_t shmem = (size_t)SMEMF * sizeof(float);   // 298,944 B < 320 KB
  (void)hipFuncSetAttribute((const void*)fused_conv_net,
                            hipFuncAttributeMaxDynamicSharedMemorySize,
                            (int)shmem);

  dim3 grid(NB / NS);   // 512 workgroups of 16 samples
  dim3 block(NT);       // 8 wave32
  fused_conv_net<<<grid, block, shmem, stream>>>(
      x, wA1, bA1, wA2, bA2, wA3, bA3, wA4, bA4, wA5, bA5,
      wB1, bB1, wB2, bB2, wC, bC, outp);
}